// ContrastiveTokenRepresentations_1125281431620
// MI455X (gfx1250) — compile-verified
//
#include <hip/hip_runtime.h>
#include <hip/hip_bf16.h>

// Problem dims (from reference)
#define BATCH   4
#define SEQ     2048
#define VOCAB   50257
#define NB      32
#define ROWS    (BATCH * SEQ)          // 8192
#define MTILES  (ROWS / 16)            // 512
#define KCHUNK  4096
#define NKCH    ((VOCAB + KCHUNK - 1) / KCHUNK)   // 13
#define INV_TEMP_DIV 0.07f

typedef __attribute__((ext_vector_type(2))) float v2f;
typedef __attribute__((ext_vector_type(8))) float v8f;

// ---------------------------------------------------------------------------
// Kernel 0: zero the sims accumulator in workspace (ws is poisoned by harness)
// ---------------------------------------------------------------------------
__global__ void zero_ws_kernel(float* __restrict__ p, int n) {
    int i = blockIdx.x * blockDim.x + threadIdx.x;
    if (i < n) p[i] = 0.0f;
}

// ---------------------------------------------------------------------------
// Kernel 1: sims[8192,32] = onehot[8192,V] @ prototypes[32,V]^T  via fp32 WMMA
// One wave32 per block. blockIdx.x = K-chunk, blockIdx.y = M-tile (16 rows).
// A fragment (16x4 f32): lane L holds M = L&15, K = (L>>4)*2 + {0,1}
// B fragment (4x16 f32): lane L holds N = L&15, K = (L>>4)*2 + {0,1}
// C/D (16x16 f32): vgpr j, lane L holds M = (L>>4)*8 + j, N = L&15
// ---------------------------------------------------------------------------
__global__ __launch_bounds__(32)
void sims_wmma_kernel(const float* __restrict__ onehot,
                      const float* __restrict__ proto,
                      float* __restrict__ sims) {
    const int lane  = threadIdx.x;          // 0..31
    const int mtile = blockIdx.y;           // 0..MTILES-1
    const int kc    = blockIdx.x;           // 0..NKCH-1
    const int m     = lane & 15;            // A row within tile / B column
    const int kq    = lane >> 4;            // 0 or 1 (K sub-pair)

    const int kstart = kc * KCHUNK;
    const int kend   = (kstart + KCHUNK < VOCAB) ? (kstart + KCHUNK) : VOCAB;

    const long long aBase  = (long long)(mtile * 16 + m) * VOCAB; // A row
    const long long b0Base = (long long)m * VOCAB;                // proto row n
    const long long b1Base = (long long)(m + 16) * VOCAB;         // proto row n+16

    // Per-lane stepping pointers (byte-bump form: minimal per-iter VALU work,
    // lets the unroller use immediate offsets within each 4-iter group).
    const float* __restrict__ aPtr  = onehot + aBase  + kstart + kq * 2;
    const float* __restrict__ bPtr0 = proto  + b0Base + kstart + kq * 2;
    const float* __restrict__ bPtr1 = proto  + b1Base + kstart + kq * 2;

    v8f c0 = {};   // N = 0..15
    v8f c1 = {};   // N = 16..31

    const int nIter = (kend - kstart) >> 2;   // full K=4 steps

    #pragma unroll 4
    for (int it = 0; it < nIter; ++it) {
        // A: streamed exactly once -> non-temporal, keep it out of L2.
        // Only 4B-aligned (row stride V is odd); in-row element offset = kstart + 4*it + kq*2 (even).
        v2f a = __builtin_nontemporal_load(reinterpret_cast<const v2f*>(aPtr));
        // B: prototype rows have odd stride V -> only 4B-guaranteed alignment;
        // scalar loads (compiler merges where legal), served from L1/L2.
        v2f bA, bB;
        bA.x = bPtr0[0];
        bA.y = bPtr0[1];
        bB.x = bPtr1[0];
        bB.y = bPtr1[1];

        c0 = __builtin_amdgcn_wmma_f32_16x16x4_f32(false, a, false, bA,
                                                   (short)0, c0, false, false);
        c1 = __builtin_amdgcn_wmma_f32_16x16x4_f32(false, a, false, bB,
                                                   (short)0, c1, false, false);
        aPtr  += 4;
        bPtr0 += 4;
        bPtr1 += 4;
    }

    const int kfull = kstart + (nIter << 2);
    if (kfull < kend) {                      // only last chunk: V % 4 == 1
        const int klo = kfull + kq * 2;
        v2f a, bA, bB;
        a.x  = (klo     < kend) ? onehot[aBase + klo]      : 0.0f;
        a.y  = (klo + 1 < kend) ? onehot[aBase + klo + 1]  : 0.0f;
        bA.x = (klo     < kend) ? proto[b0Base + klo]      : 0.0f;
        bA.y = (klo + 1 < kend) ? proto[b0Base + klo + 1]  : 0.0f;
        bB.x = (klo     < kend) ? proto[b1Base + klo]      : 0.0f;
        bB.y = (klo + 1 < kend) ? proto[b1Base + klo + 1]  : 0.0f;

        c0 = __builtin_amdgcn_wmma_f32_16x16x4_f32(false, a, false, bA,
                                                   (short)0, c0, false, false);
        c1 = __builtin_amdgcn_wmma_f32_16x16x4_f32(false, a, false, bB,
                                                   (short)0, c1, false, false);
    }

    // Scatter-accumulate the partial 16x32 tile. Partials are exact zeros
    // except the chunk holding the hot index, so order is irrelevant.
    #pragma unroll
    for (int j = 0; j < 8; ++j) {
        const int row = mtile * 16 + kq * 8 + j;
        atomicAdd(&sims[row * NB + m],       c0[j]);
        atomicAdd(&sims[row * NB + m + 16],  c1[j]);
    }
}

// ---------------------------------------------------------------------------
// Kernel 2: per-row gumbel-softmax(hard=True) straight-through output.
// One wave32 per row, lane n handles channel n (NB == 32 == wave width).
// ---------------------------------------------------------------------------
__global__ __launch_bounds__(256)
void gumbel_hard_kernel(const float* __restrict__ sims,
                        const float* __restrict__ gumbel,
                        float* __restrict__ out) {
    const int row  = (blockIdx.x * blockDim.x + threadIdx.x) >> 5; // exact grid
    const int lane = threadIdx.x & 31;
    const int idx  = row * NB + lane;

    // logits = sims / TEMP; z = (logits + gumbel) / tau, tau == 1
    const float z = sims[idx] / INV_TEMP_DIV + gumbel[idx];

    // max-reduce for numerically stable softmax
    float mx = z;
    #pragma unroll
    for (int off = 16; off > 0; off >>= 1)
        mx = fmaxf(mx, __shfl_xor(mx, off, 32));

    const float e = expf(z - mx);
    float s = e;
    #pragma unroll
    for (int off = 16; off > 0; off >>= 1)
        s += __shfl_xor(s, off, 32);

    const float y = e / s;   // y_soft

    // argmax with first-occurrence tie-break (matches jnp.argmax)
    float bv = z;
    int   bi = lane;
    #pragma unroll
    for (int off = 16; off > 0; off >>= 1) {
        const float ov = __shfl_xor(bv, off, 32);
        const int   oi = __shfl_xor(bi, off, 32);
        if (ov > bv || (ov == bv && oi < bi)) { bv = ov; bi = oi; }
    }

    const float hard = (lane == bi) ? 1.0f : 0.0f;
    // reference evaluation order: (y_hard - y_soft) + y_soft
    out[idx] = (hard - y) + y;
}

// ---------------------------------------------------------------------------
extern "C" void kernel_launch(void* const* d_in, const int* in_sizes, int n_in,
                              void* d_out, int out_size, void* d_ws, size_t ws_size,
                              hipStream_t stream) {
    const float* onehot = (const float*)d_in[0];   // [B,S,V] fp32
    const float* proto  = (const float*)d_in[1];   // [NB,V]  fp32
    const float* gumbel = (const float*)d_in[2];   // [B,S,NB] fp32
    float*       out    = (float*)d_out;           // [B,S,NB] fp32
    float*       sims   = (float*)d_ws;            // [ROWS,NB] fp32 accumulator

    const int nsims = ROWS * NB;
    zero_ws_kernel<<<(nsims + 255) / 256, 256, 0, stream>>>(sims, nsims);

    dim3 grid(NKCH, MTILES);
    sims_wmma_kernel<<<grid, 32, 0, stream>>>(onehot, proto, sims);

    gumbel_hard_kernel<<<ROWS / 8, 256, 0, stream>>>(sims, gumbel, out);
}